// Loss_25546465477236
// MI455X (gfx1250) — compile-verified
//
#include <hip/hip_runtime.h>
#include <hip/hip_bf16.h>

typedef __attribute__((ext_vector_type(2))) float v2f;
typedef __attribute__((ext_vector_type(8))) float v8f;

#define S_FEAT 80.0f
#define LAMBDA_COORD 5.0f
#define LAMBDA_NOOBJ 0.5f

// Full wave32 sum via chained V_WMMA_F32_16X16X4_F32.
// Stage 1: A[m][:] = [p_m, 0, p_{m+16}, 0], B = ones  ->  D1[m][n] = p_m + p_{m+16} (all n).
// Stage 2: A = ones, B = (D1.v2j, D1.v2j+1) pairs, accumulated over j=0..3:
//          each WMMA adds rows {2j, 2j+1, 8+2j, 9+2j} of D1; union = all 16 rows.
// Result: every element of acc == sum over all 32 lanes.
__device__ __forceinline__ float wave_reduce_wmma(float p) {
    v2f a;    a[0] = p;    a[1] = 0.0f;
    v2f ones; ones[0] = 1.0f; ones[1] = 1.0f;
    v8f zero = {};
    v8f d1 = __builtin_amdgcn_wmma_f32_16x16x4_f32(
        false, a, false, ones, (short)0, zero, false, false);

    v2f b01; b01[0] = d1[0]; b01[1] = d1[1];
    v2f b23; b23[0] = d1[2]; b23[1] = d1[3];
    v2f b45; b45[0] = d1[4]; b45[1] = d1[5];
    v2f b67; b67[0] = d1[6]; b67[1] = d1[7];

    v8f acc = {};
    acc = __builtin_amdgcn_wmma_f32_16x16x4_f32(false, ones, false, b01, (short)0, acc, false, false);
    acc = __builtin_amdgcn_wmma_f32_16x16x4_f32(false, ones, false, b23, (short)0, acc, false, false);
    acc = __builtin_amdgcn_wmma_f32_16x16x4_f32(false, ones, false, b45, (short)0, acc, false, false);
    acc = __builtin_amdgcn_wmma_f32_16x16x4_f32(false, ones, false, b67, (short)0, acc, false, false);
    return acc[0];
}

__device__ __forceinline__ float cell_loss(float px, float py, float pw, float ph, float pc,
                                           float tx, float ty, float tw, float th, int mk) {
    const float m = (mk > 0) ? 1.0f : 0.0f;

    // _xyxy (faithful to reference bug: center for both x and y is w/S)
    const float pcn = pw * (1.0f / S_FEAT);
    const float px1 = pcn - 0.5f * pw;
    const float py1 = pcn - 0.5f * ph;
    const float px2 = pcn + 0.5f * pw;
    const float py2 = pcn + 0.5f * ph;

    const float tcn = tw * (1.0f / S_FEAT);
    const float tx1 = tcn - 0.5f * tw;
    const float ty1 = tcn - 0.5f * th;
    const float tx2 = tcn + 0.5f * tw;
    const float ty2 = tcn + 0.5f * th;

    const float iw = fmaxf(fminf(px2, tx2) - fmaxf(px1, tx1), 0.0f);
    const float ih = fmaxf(fminf(py2, ty2) - fmaxf(py1, ty1), 0.0f);
    const float inter = iw * ih;
    const float uni   = pw * ph + tw * th - inter;
    const float iou   = inter / uni;

    const float dx = px - tx;
    const float dy = py - ty;
    const float dw = sqrtf(pw) - sqrtf(tw);
    const float dh = sqrtf(ph) - sqrtf(th);
    const float dc = pc - iou;

    const float loss_xy_wh = dx * dx + dy * dy + dw * dw + dh * dh;
    return m * (LAMBDA_COORD * loss_xy_wh + dc * dc)
         + (1.0f - m) * (LAMBDA_NOOBJ * pc * pc);
}

__global__ void yolo_loss_partial_kernel(const float* __restrict__ pred,
                                         const float* __restrict__ tgt,
                                         const int*  __restrict__ mask,
                                         float* __restrict__ block_partials,
                                         long long ncells) {
    const int tid           = threadIdx.x;
    const long long gtid    = (long long)blockIdx.x * blockDim.x + tid;
    const long long gstride = (long long)gridDim.x * blockDim.x;
    const long long ngroups = ncells >> 2;   // 4 cells per thread-iteration

    const float4* __restrict__ p4 = reinterpret_cast<const float4*>(pred);
    const float4* __restrict__ t4 = reinterpret_cast<const float4*>(tgt);
    const int4*   __restrict__ m4 = reinterpret_cast<const int4*>(mask);

    float p = 0.0f;  // per-lane partial loss

    for (long long g = gtid; g < ngroups; g += gstride) {
        // 5x GLOBAL_LOAD_B128: 20 pred floats = cells [4g, 4g+4)
        const float4 r0 = p4[g * 5 + 0];
        const float4 r1 = p4[g * 5 + 1];
        const float4 r2 = p4[g * 5 + 2];
        const float4 r3 = p4[g * 5 + 3];
        const float4 r4 = p4[g * 5 + 4];
        // 4x GLOBAL_LOAD_B128: 16 target floats
        const float4 s0 = t4[g * 4 + 0];
        const float4 s1 = t4[g * 4 + 1];
        const float4 s2 = t4[g * 4 + 2];
        const float4 s3 = t4[g * 4 + 3];
        // 1x GLOBAL_LOAD_B128: 4 mask ints
        const int4 mm = m4[g];

        const float pf[20] = { r0.x, r0.y, r0.z, r0.w,
                               r1.x, r1.y, r1.z, r1.w,
                               r2.x, r2.y, r2.z, r2.w,
                               r3.x, r3.y, r3.z, r3.w,
                               r4.x, r4.y, r4.z, r4.w };
        const float tf[16] = { s0.x, s0.y, s0.z, s0.w,
                               s1.x, s1.y, s1.z, s1.w,
                               s2.x, s2.y, s2.z, s2.w,
                               s3.x, s3.y, s3.z, s3.w };
        const int   mi[4]  = { mm.x, mm.y, mm.z, mm.w };

        #pragma unroll
        for (int c = 0; c < 4; ++c) {
            p += cell_loss(pf[5 * c + 0], pf[5 * c + 1], pf[5 * c + 2],
                           pf[5 * c + 3], pf[5 * c + 4],
                           tf[4 * c + 0], tf[4 * c + 1],
                           tf[4 * c + 2], tf[4 * c + 3],
                           mi[c]);
        }
    }

    // scalar tail for ncells not divisible by 4
    for (long long i = (ngroups << 2) + gtid; i < ncells; i += gstride) {
        const long long b5 = i * 5;
        const long long b4 = i * 4;
        p += cell_loss(pred[b5 + 0], pred[b5 + 1], pred[b5 + 2],
                       pred[b5 + 3], pred[b5 + 4],
                       tgt[b4 + 0], tgt[b4 + 1], tgt[b4 + 2], tgt[b4 + 3],
                       mask[i]);
    }

    // --- wave reduction (full EXEC: reconverged after the loops) ---
    const float wsum = wave_reduce_wmma(p);

    __shared__ float s_wave[8];              // 256 threads = 8 wave32
    const int lane = tid & 31;
    const int wave = tid >> 5;
    if (lane == 0) s_wave[wave] = wsum;
    __syncthreads();

    if (wave == 0) {                         // wave-uniform branch: EXEC stays all-1s
        float q = (lane < 8) ? s_wave[lane] : 0.0f;
        const float bsum = wave_reduce_wmma(q);
        if (lane == 0) block_partials[blockIdx.x] = bsum;
    }
}

__global__ void yolo_loss_finalize_kernel(const float* __restrict__ block_partials,
                                          float* __restrict__ out,
                                          int nblocks, float inv_batch) {
    if (threadIdx.x == 0 && blockIdx.x == 0) {
        double acc = 0.0;
        for (int i = 0; i < nblocks; ++i) acc += (double)block_partials[i];
        out[0] = (float)(acc * (double)inv_batch);
    }
}

extern "C" void kernel_launch(void* const* d_in, const int* in_sizes, int n_in,
                              void* d_out, int out_size, void* d_ws, size_t ws_size,
                              hipStream_t stream) {
    const float* pred = (const float*)d_in[0];   // [N,80,80,5] f32
    const float* tgt  = (const float*)d_in[1];   // [N,80,80,4] f32
    const int*   msk  = (const int*)d_in[2];     // [N,80,80] i32
    float* out = (float*)d_out;
    float* ws  = (float*)d_ws;

    const long long ncells = (long long)in_sizes[2];      // N*80*80
    const long long batch  = ncells / (80LL * 80LL);

    const int threads = 256;
    long long want = (ncells / 4 + threads - 1) / threads;
    int blocks = (int)((want < 2048) ? (want > 0 ? want : 1) : 2048);
    // keep workspace usage bounded (blocks * 4 bytes)
    if ((size_t)blocks * sizeof(float) > ws_size && ws_size >= sizeof(float))
        blocks = (int)(ws_size / sizeof(float));

    yolo_loss_partial_kernel<<<blocks, threads, 0, stream>>>(pred, tgt, msk, ws, ncells);
    yolo_loss_finalize_kernel<<<1, 32, 0, stream>>>(ws, out, blocks,
                                                    1.0f / (float)batch);
}